// NT_Xent_16930761081030
// MI455X (gfx1250) — compile-verified
//
#include <hip/hip_runtime.h>
#include <math.h>

// Problem constants (match reference)
#define BB   8192
#define DD   512
#define KK   2048
#define EPSV 1e-8f
#define INVT 2.0f   // 1 / TEMP, TEMP = 0.5

typedef __attribute__((ext_vector_type(2))) float v2f;  // WMMA f32 A/B operand
typedef __attribute__((ext_vector_type(8))) float v8f;  // WMMA f32 C/D operand

// ---------------------------------------------------------------------------
// gfx1250 async global->LDS copy (ASYNCcnt path). One b64 per lane lets us do
// the WMMA-operand swizzle during the copy itself (a rectangular TDM tile
// could not land data in this conflict-free layout).
// ---------------------------------------------------------------------------
__device__ __forceinline__ void asyncCopyB64(unsigned ldsByteAddr,
                                             const float* __restrict__ base,
                                             int byteOff) {
  asm volatile("global_load_async_to_lds_b64 %0, %1, %2"
               :: "v"(ldsByteAddr), "v"(byteOff), "s"(base)
               : "memory");
}
__device__ __forceinline__ void waitAsync() {
  asm volatile("s_wait_asynccnt 0x0" ::: "memory");
}
__device__ __forceinline__ unsigned ldsAddr(const void* p) {
  return (unsigned)(uintptr_t)p;   // low 32 bits of a generic LDS pointer = LDS offset
}

// ---------------------------------------------------------------------------
// Kernel 1: per-row L2 norms of z_i / z_j and the positive logit
// ---------------------------------------------------------------------------
__global__ void __launch_bounds__(128) prep_norms(
    const float* __restrict__ zi, const float* __restrict__ zj,
    float* __restrict__ ni, float* __restrict__ nj, float* __restrict__ poslog) {
  const int wave = threadIdx.x >> 5;
  const int lane = threadIdx.x & 31;
  const int row  = blockIdx.x * 4 + wave;
  const float4* a = (const float4*)(zi + (size_t)row * DD);
  const float4* b = (const float4*)(zj + (size_t)row * DD);
  float si = 0.f, sj = 0.f, sd = 0.f;
#pragma unroll
  for (int t = 0; t < DD / 4; t += 32) {
    float4 x = a[t + lane];
    float4 y = b[t + lane];
    si += x.x * x.x + x.y * x.y + x.z * x.z + x.w * x.w;
    sj += y.x * y.x + y.y * y.y + y.z * y.z + y.w * y.w;
    sd += x.x * y.x + x.y * y.y + x.z * y.z + x.w * y.w;
  }
#pragma unroll
  for (int m = 16; m; m >>= 1) {
    si += __shfl_xor(si, m);
    sj += __shfl_xor(sj, m);
    sd += __shfl_xor(sd, m);
  }
  if (lane == 0) {
    float na = sqrtf(si), nb = sqrtf(sj);
    ni[row] = na;
    nj[row] = nb;
    poslog[row] = sd / fmaxf(na * nb, EPSV) * INVT;
  }
}

// ---------------------------------------------------------------------------
// Kernel 2: gather negatives: G0 = z_j[perm], G1 = z_j[perm+1] (8 MB, L2-
// resident), plus their norms. col = p + (p>=i) only ever needs these two.
// ---------------------------------------------------------------------------
__global__ void __launch_bounds__(128) gather_negs(
    const float* __restrict__ zj, const int* __restrict__ perm,
    const float* __restrict__ nj,
    float* __restrict__ g0, float* __restrict__ g1,
    float* __restrict__ njg0, float* __restrict__ njg1) {
  const int k = blockIdx.x;
  const int p = perm[k];
  const float4* s0 = (const float4*)(zj + (size_t)p * DD);
  const float4* s1 = (const float4*)(zj + (size_t)(p + 1) * DD);
  float4* d0 = (float4*)(g0 + (size_t)k * DD);
  float4* d1 = (float4*)(g1 + (size_t)k * DD);
  const int t = threadIdx.x;
  d0[t] = s0[t];
  d1[t] = s1[t];
  if (t == 0) {
    njg0[k] = nj[p];
    njg1[k] = nj[p + 1];
  }
}

// ---------------------------------------------------------------------------
// Kernel 3: fused double-GEMM + online logsumexp/rank.
//
// Block = 128 threads (4 waves) owning 64 rows (16 per wave). All 4 waves
// walk the SAME column tile together so each 16-col B slab is fetched once
// per block (4x less L2 traffic) into double-buffered LDS via async copies.
//
// LDS layout is WMMA-operand-major: tile stored as [d-pair q][16 cols] of
// float2. Operand fetch = one ds_load_b64/lane; lanes 0-15 hit banks 2l,2l+1
// and lanes 16-31 (q+1 -> +32 words) hit banks 32+2l -> conflict-free.
//   A (16x4): lane<16 holds (K=0,K=1) of row lane; lane>=16 holds (K=2,K=3)
//   B (4x16) mirrors it; C/D: VGPR v = rows v / v+8 per half-wave.
// ---------------------------------------------------------------------------
__global__ void __launch_bounds__(128) nt_xent_main(
    const float* __restrict__ zi, const int* __restrict__ perm,
    const float* __restrict__ g0, const float* __restrict__ g1,
    const float* __restrict__ ni, const float* __restrict__ njg0,
    const float* __restrict__ njg1, const float* __restrict__ poslog,
    float* __restrict__ rowLoss, int* __restrict__ rowCnt) {
  // A panel: 4 row-groups x 256 d-pairs x 16 rows (128 KB)
  __shared__ float2 A_s[4 * 256 * 16];
  // B slabs: [parity][mat*64 + q][16 cols] (2 x 16 KB)
  __shared__ float2 B_s[2][2 * 64 * 16];

  const int t    = threadIdx.x;
  const int wave = t >> 5;
  const int lane = t & 31;
  const int l15  = lane & 15;
  const int h    = lane >> 4;          // half-wave: K-pair select
  const int hi8  = h << 3;             // 0 or 8 : row offset in C/D
  const int rblk = blockIdx.x * 64;
  const int r    = rblk + wave * 16;   // this wave's 16-row stripe

  // ---- issue async fill of the whole A panel (swizzled) ------------------
  {
    const unsigned aBase = ldsAddr(&A_s[0]);
#pragma unroll
    for (int j = 0; j < 128; ++j) {
      const int idx   = t + 128 * j;         // 0..16383
      const int col   = idx & 15;            // row within group
      const int group = (idx >> 4) & 3;
      const int q     = idx >> 6;            // 0..255 (d-pair)
      const int goff  = ((rblk + group * 16 + col) * DD + 2 * q) * 4;
      asyncCopyB64(aBase + (unsigned)(((group * 256 + q) * 16 + col) * 8),
                   zi, goff);
    }
  }
  // ---- issue async fill of B slab 0 (ct=0, dslab=0) ----------------------
  const unsigned bBase0 = ldsAddr(&B_s[0][0]);
  const unsigned bBase1 = ldsAddr(&B_s[1][0]);
#define FILL_B_SLAB(CT, DSLAB, BBASE)                                        \
  do {                                                                       \
    const int k0_ = (CT) * 16, db_ = (DSLAB) * 128;                          \
    _Pragma("unroll")                                                        \
    for (int j = 0; j < 16; ++j) {                                           \
      const int mat  = j >> 3;            /* compile-time: uniform src */    \
      const int idxr = t + 128 * (j & 7); /* 0..1023 */                      \
      const int col  = idxr & 15;                                            \
      const int q    = idxr >> 4;         /* 0..63 */                        \
      const float* src = mat ? g1 : g0;                                      \
      const int goff = ((k0_ + col) * DD + db_ + 2 * q) * 4;                 \
      asyncCopyB64((BBASE) + (unsigned)(((mat * 64 + q) * 16 + col) * 8),    \
                   src, goff);                                               \
    }                                                                        \
  } while (0)

  FILL_B_SLAB(0, 0, bBase0);
  waitAsync();
  __syncthreads();

  // ---- per-lane row metadata --------------------------------------------
  float niv[8], posv[8], se[8];
  int cnt[8];
#pragma unroll
  for (int v = 0; v < 8; ++v) {
    const int i = r + v + hi8;
    niv[v]  = ni[i];
    posv[v] = poslog[i];
    se[v]   = 0.f;
    cnt[v]  = 0;
  }

  const float2* apan = &A_s[wave * 4096];   // this wave's swizzled A panel

  for (int ct = 0; ct < KK / 16; ++ct) {
    const int k = ct * 16 + l15;            // this lane's negative column
    const int p = perm[k];
    const float nj0 = njg0[k];
    const float nj1 = njg1[k];
    v8f acc0 = {0, 0, 0, 0, 0, 0, 0, 0};
    v8f acc1 = {0, 0, 0, 0, 0, 0, 0, 0};

    for (int dslab = 0; dslab < 4; ++dslab) {
      const int s   = ct * 4 + dslab;
      const int par = s & 1;
      // prefetch next slab into the other buffer while computing this one
      if (s + 1 < 512) {
        const int nct = (s + 1) >> 2, nds = (s + 1) & 3;
        if (par) FILL_B_SLAB(nct, nds, bBase0);
        else     FILL_B_SLAB(nct, nds, bBase1);
      }
      const float2* bp = &B_s[par][0];
      const int dbase = dslab * 128;
#pragma unroll
      for (int dloc = 0; dloc < 128; dloc += 4) {
        const int qa = ((dbase + dloc) >> 1) + h;  // A d-pair (global)
        const int qb = (dloc >> 1) + h;            // B d-pair (slab-local)
        const float2 avp = apan[qa * 16 + l15];
        const float2 b0p = bp[qb * 16 + l15];
        const float2 b1p = bp[1024 + qb * 16 + l15];
        const v2f av = {avp.x, avp.y};
        const v2f b0 = {b0p.x, b0p.y};
        const v2f b1 = {b1p.x, b1p.y};
        acc0 = __builtin_amdgcn_wmma_f32_16x16x4_f32(false, av, false, b0,
                                                     (short)0, acc0, false, false);
        acc1 = __builtin_amdgcn_wmma_f32_16x16x4_f32(false, av, false, b1,
                                                     (short)0, acc1, false, false);
      }
      waitAsync();        // prefetch for s+1 has landed
      __syncthreads();    // nobody still reads buffer par before it's refilled
    }

    // Epilogue: diagonal-shift select + normalize + online accumulation
#pragma unroll
    for (int v = 0; v < 8; ++v) {
      const int   i   = r + v + hi8;
      const bool  ge  = (p >= i);               // col = p+1 ? (G1) : (G0)
      const float dot = ge ? acc1[v] : acc0[v];
      const float njs = ge ? nj1 : nj0;
      const float sim = dot / fmaxf(niv[v] * njs, EPSV) * INVT;
      se[v]  += expf(sim);
      cnt[v] += (sim > posv[v]) ? 1 : 0;
    }
  }

  // Reduce each row across the 16 lanes that share it (masks<16 stay within
  // a half-wave on wave32), then write per-row results.
#pragma unroll
  for (int v = 0; v < 8; ++v) {
    float s = se[v];
    int   c = cnt[v];
#pragma unroll
    for (int m = 8; m; m >>= 1) {
      s += __shfl_xor(s, m);
      c += __shfl_xor(c, m);
    }
    if (l15 == 0) {
      const int i = r + v + hi8;
      const float tot = s + expf(posv[v]);     // include the positive logit
      rowLoss[i] = logf(tot) - posv[v];        // |sim|<=4 -> no max pass needed
      rowCnt[i]  = c;
    }
  }
#undef FILL_B_SLAB
}

// ---------------------------------------------------------------------------
// Kernel 4: deterministic tree reduction -> (loss, acc1, acc5)
// ---------------------------------------------------------------------------
__global__ void __launch_bounds__(256) finalize(
    const float* __restrict__ rowLoss, const int* __restrict__ rowCnt,
    float* __restrict__ out) {
  __shared__ float sl[256];
  __shared__ int   s1[256];
  __shared__ int   s5[256];
  const int t = threadIdx.x;
  float L = 0.f;
  int a1 = 0, a5 = 0;
  for (int i = t; i < BB; i += 256) {
    L += rowLoss[i];
    const int c = rowCnt[i];
    a1 += (c == 0) ? 1 : 0;
    a5 += (c < 5) ? 1 : 0;
  }
  sl[t] = L; s1[t] = a1; s5[t] = a5;
  __syncthreads();
  for (int s = 128; s > 0; s >>= 1) {
    if (t < s) {
      sl[t] += sl[t + s];
      s1[t] += s1[t + s];
      s5[t] += s5[t + s];
    }
    __syncthreads();
  }
  if (t == 0) {
    out[0] = sl[0] / (float)BB;
    out[1] = 100.f * (float)s1[0] / (float)BB;
    out[2] = 100.f * (float)s5[0] / (float)BB;
  }
}

// ---------------------------------------------------------------------------
extern "C" void kernel_launch(void* const* d_in, const int* in_sizes, int n_in,
                              void* d_out, int out_size, void* d_ws, size_t ws_size,
                              hipStream_t stream) {
  const float* zi   = (const float*)d_in[0];
  const float* zj   = (const float*)d_in[1];
  const int*   perm = (const int*)d_in[2];
  float* out = (float*)d_out;

  // Workspace carve-up (~8.6 MB)
  float* w      = (float*)d_ws;
  float* g0     = w; w += (size_t)KK * DD;
  float* g1     = w; w += (size_t)KK * DD;
  float* ni     = w; w += BB;
  float* nj     = w; w += BB;
  float* njg0   = w; w += KK;
  float* njg1   = w; w += KK;
  float* poslog = w; w += BB;
  float* rowLoss = w; w += BB;
  int*   rowCnt  = (int*)w;

  prep_norms <<<BB / 4, 128, 0, stream>>>(zi, zj, ni, nj, poslog);
  gather_negs<<<KK,     128, 0, stream>>>(zj, perm, nj, g0, g1, njg0, njg1);
  nt_xent_main<<<BB / 64, 128, 0, stream>>>(zi, perm, g0, g1, ni, njg0, njg1,
                                            poslog, rowLoss, rowCnt);
  finalize<<<1, 256, 0, stream>>>(rowLoss, rowCnt, out);
}